// FDN_88012469830490
// MI455X (gfx1250) — compile-verified
//
#include <hip/hip_runtime.h>
#include <math.h>

// ---------------- constants ----------------
static const int kL   = 88200;     // IR length
static const int kFB  = 44101;     // rfft bins of L
static const int kNB  = 1 << 18;   // Bluestein FFT size (>= 2L-1), = 4^9
static const int kST_NB = 9;       // radix-4 stages
static const int kNCV = 1 << 20;   // conv FFT size (>= T+L-1 = 529199), = 4^10
static const int kST_NCV = 10;     // radix-4 stages
static const int kT   = 441000;    // signal length
static const int kBAT = 16;        // batch
static const int kGRP = 4;         // signals per L2-resident FFT group (64 MB working set)

#define TWO_PI_D 6.283185307179586476925287

__constant__ int g_delays[6] = {997, 1153, 1327, 1559, 1801, 2099};

typedef float wv2 __attribute__((ext_vector_type(2)));
typedef float wv8 __attribute__((ext_vector_type(8)));

__device__ __forceinline__ float2 cmulf(float2 a, float2 b) {
  return make_float2(a.x * b.x - a.y * b.y, a.x * b.y + a.y * b.x);
}

// ================= WMMA f32 16x16 matmul on LDS tiles =================
// C/D layout: VGPR v, lanes0-15 -> M=v,N=lane; lanes16-31 -> M=v+8.
// A 16x4: lane -> M=lane%16, VGPR j -> K = j + 2*(lane/16).
// B 4x16 mirrored: VGPR j -> K = j + 2*(lane/16), N = lane%16.
__device__ __forceinline__ void mm16(const float* A, const float* Bm, float* D, int lane) {
  const int row = lane & 15;
  const int hi  = lane >> 4;
  wv8 c = {0.f, 0.f, 0.f, 0.f, 0.f, 0.f, 0.f, 0.f};
#pragma unroll
  for (int q = 0; q < 4; ++q) {
    const int k0 = q * 4 + hi * 2;
    wv2 a, b;
    a[0] = A[row * 16 + k0];
    a[1] = A[row * 16 + k0 + 1];
    b[0] = Bm[k0 * 16 + row];
    b[1] = Bm[(k0 + 1) * 16 + row];
    c = __builtin_amdgcn_wmma_f32_16x16x4_f32(false, a, false, b, (short)0, c, false, false);
  }
#pragma unroll
  for (int v = 0; v < 8; ++v) D[(v + hi * 8) * 16 + row] = c[v];
}

// expm(skew(U_raw)) via scaling-and-squaring Taylor, 6x6 padded to 16x16.
// Also builds gamma table = sigmoid(g_raw) * GAMMA_MAX. One wave.
__global__ void k_expm(const float* __restrict__ U_raw,
                       const float* __restrict__ g_raw,
                       float* __restrict__ U_out,
                       float* __restrict__ G_out) {
  __shared__ float S[256], P[256], T[256], TMP[256];
  const int lane = threadIdx.x;
  for (int idx = lane; idx < 256; idx += 32) {
    int r = idx >> 4, c = idx & 15;
    float v = 0.f;
    if (r < 6 && c < 6) {
      float up = (c > r) ? U_raw[r * 6 + c] : 0.f;
      float lo = (r > c) ? U_raw[c * 6 + r] : 0.f;
      v = (up - lo) * (1.0f / 256.0f);   // scale by 2^-8
    }
    S[idx] = v;
    float id = (r == c) ? 1.f : 0.f;
    P[idx] = id;
    T[idx] = id;
  }
  __syncthreads();
  for (int k = 1; k <= 12; ++k) {        // Taylor terms
    mm16(P, S, TMP, lane);
    __syncthreads();
    float inv = 1.f / (float)k;
    for (int idx = lane; idx < 256; idx += 32) {
      float pv = TMP[idx] * inv;
      P[idx] = pv;
      T[idx] += pv;
    }
    __syncthreads();
  }
  for (int t = 0; t < 8; ++t) {          // square back 8 times
    mm16(T, T, TMP, lane);
    __syncthreads();
    for (int idx = lane; idx < 256; idx += 32) T[idx] = TMP[idx];
    __syncthreads();
  }
  for (int idx = lane; idx < 36; idx += 32)
    U_out[idx] = T[(idx / 6) * 16 + (idx % 6)];
  for (int idx = lane; idx < 49 * 6; idx += 32) {
    int d = idx % 6;
    float gm = __powf(10.f, -(float)g_delays[d] * (1.f / 22050.f));  // GAMMA_MAX
    G_out[idx] = gm / (1.f + __expf(-g_raw[idx]));
  }
}

// ================= per-bin 6x6 complex solve -> H (2x2 per bin) =================
__global__ void k_solve(const float* __restrict__ U, const float* __restrict__ G,
                        const float* __restrict__ bmat, const float* __restrict__ cmat,
                        float2* __restrict__ Hs) {
  int f = blockIdx.x * blockDim.x + threadIdx.x;
  if (f >= kFB) return;
  // gamma interp (exact rational arithmetic, align_corners)
  int num = f * 48;                     // f * (NDF-1)
  int i0 = num / 44100;                 // / (FB-1)
  if (i0 > 47) i0 = 47;
  float frac = (float)(num - i0 * 44100) * (1.f / 44100.f);
  float g[6];
#pragma unroll
  for (int d = 0; d < 6; ++d)
    g[d] = G[i0 * 6 + d] * (1.f - frac) + G[(i0 + 1) * 6 + d] * frac;

  float Mr[6][6], Mi[6][6], Xr[6][2], Xi[6][2];
#pragma unroll
  for (int r = 0; r < 6; ++r)
#pragma unroll
    for (int c = 0; c < 6; ++c) { Mr[r][c] = -U[r * 6 + c] * g[c]; Mi[r][c] = 0.f; }
#pragma unroll
  for (int d = 0; d < 6; ++d) {
    long long rm = ((long long)f * (long long)g_delays[d]) % (long long)kL; // exact phase mod
    float ang = (float)(TWO_PI_D * (double)rm / (double)kL);
    float sn, cs; __sincosf(ang, &sn, &cs);
    Mr[d][d] += cs;  Mi[d][d] += sn;   // e^{+i w delay}
  }
#pragma unroll
  for (int i = 0; i < 6; ++i)
#pragma unroll
    for (int o = 0; o < 2; ++o) { Xr[i][o] = bmat[i * 2 + o]; Xi[i][o] = 0.f; }

  // Gauss-Jordan (no pivot; diag ~ unit-modulus dominated)
#pragma unroll
  for (int p = 0; p < 6; ++p) {
    float dr = Mr[p][p], di = Mi[p][p];
    float den = 1.f / (dr * dr + di * di);
    float ir = dr * den, ii = -di * den;
#pragma unroll
    for (int c = 0; c < 6; ++c) if (c > p) {
      float tr = Mr[p][c] * ir - Mi[p][c] * ii;
      float ti = Mr[p][c] * ii + Mi[p][c] * ir;
      Mr[p][c] = tr; Mi[p][c] = ti;
    }
#pragma unroll
    for (int o = 0; o < 2; ++o) {
      float tr = Xr[p][o] * ir - Xi[p][o] * ii;
      float ti = Xr[p][o] * ii + Xi[p][o] * ir;
      Xr[p][o] = tr; Xi[p][o] = ti;
    }
#pragma unroll
    for (int r = 0; r < 6; ++r) if (r != p) {
      float fr = Mr[r][p], fi = Mi[r][p];
#pragma unroll
      for (int c = 0; c < 6; ++c) if (c > p) {
        Mr[r][c] -= fr * Mr[p][c] - fi * Mi[p][c];
        Mi[r][c] -= fr * Mi[p][c] + fi * Mr[p][c];
      }
#pragma unroll
      for (int o = 0; o < 2; ++o) {
        Xr[r][o] -= fr * Xr[p][o] - fi * Xi[p][o];
        Xi[r][o] -= fr * Xi[p][o] + fi * Xr[p][o];
      }
    }
  }
  // H[f][co][ci] = sum_i c[co][i] * X[i][ci]
#pragma unroll
  for (int co = 0; co < 2; ++co)
#pragma unroll
    for (int ci = 0; ci < 2; ++ci) {
      float hr = 0.f, hi = 0.f;
#pragma unroll
      for (int i = 0; i < 6; ++i) {
        float cv = cmat[co * 6 + i];
        hr += cv * Xr[i][ci];
        hi += cv * Xi[i][ci];
      }
      Hs[(size_t)(co * 2 + ci) * kFB + f] = make_float2(hr, hi);
    }
}

// ================= radix-4 Stockham stage (batched, ping-pong) =================
// Port of the standard Stockham radix-4 (OTFFT form):
//   y[k + m*(4j+0)] =  (a+c) + (b+d)
//   y[k + m*(4j+1)] = ((a-c) + s*i*(b-d)) * w
//   y[k + m*(4j+2)] = ((a+c) - (b+d))     * w^2
//   y[k + m*(4j+3)] = ((a-c) - s*i*(b-d)) * w^3,  w = e^{s*2*pi*i*j/(4l)}
__global__ void k_fft4_stage(const float2* __restrict__ src, float2* __restrict__ dst,
                             int l, int mshift, float sign, int n) {
  int tid = blockIdx.x * blockDim.x + threadIdx.x;
  if (tid >= (n >> 2)) return;
  size_t base = (size_t)blockIdx.y * (size_t)n;
  int m = 1 << mshift;
  int j = tid >> mshift;
  int k = tid & (m - 1);

  float ang = sign * 1.57079632679489662f * ((float)j / (float)l);  // s*2pi*j/(4l)
  float sn, cs; __sincosf(ang, &sn, &cs);
  float2 w1 = make_float2(cs, sn);
  float2 w2 = cmulf(w1, w1);
  float2 w3 = cmulf(w2, w1);

  size_t lm = (size_t)l << mshift;
  size_t i0 = base + (size_t)k + ((size_t)j << mshift);
  float2 a = src[i0];
  float2 b = src[i0 + lm];
  float2 c = src[i0 + 2 * lm];
  float2 d = src[i0 + 3 * lm];

  float2 t0 = make_float2(a.x + c.x, a.y + c.y);
  float2 t1 = make_float2(a.x - c.x, a.y - c.y);
  float2 t2 = make_float2(b.x + d.x, b.y + d.y);
  float2 bd = make_float2(b.x - d.x, b.y - d.y);
  float2 t3 = make_float2(-sign * bd.y, sign * bd.x);     // s*i*(b-d)

  size_t o0 = base + (size_t)k + ((size_t)j << (mshift + 2));
  dst[o0]         = make_float2(t0.x + t2.x, t0.y + t2.y);
  dst[o0 + m]     = cmulf(make_float2(t1.x + t3.x, t1.y + t3.y), w1);
  dst[o0 + 2 * m] = cmulf(make_float2(t0.x - t2.x, t0.y - t2.y), w2);
  dst[o0 + 3 * m] = cmulf(make_float2(t1.x - t3.x, t1.y - t3.y), w3);
}

// ================= Bluestein pieces (inverse DFT of length L) =================
__global__ void k_blu_b(float2* __restrict__ bb) {
  int k = blockIdx.x * blockDim.x + threadIdx.x;
  if (k >= kNB) return;
  int d = (k <= kNB / 2) ? k : (kNB - k);
  float2 v = make_float2(0.f, 0.f);
  if (d < kL) {
    double t = (double)d * (double)d / (double)(2 * kL);
    t -= floor(t);
    double sn, cs; sincos(-TWO_PI_D * t, &sn, &cs);  // e^{-i pi d^2 / L}
    v = make_float2((float)cs, (float)sn);
  }
  bb[k] = v;
}

// a[m] = (S0[m] + i*S1[m]) * e^{+i pi m^2 / L}; Hermitian extension of H spectra.
__global__ void k_blu_a(const float2* __restrict__ Hs, float2* __restrict__ a) {
  int k = blockIdx.x * blockDim.x + threadIdx.x;
  int c = blockIdx.y;
  if (k >= kNB) return;
  float2 v = make_float2(0.f, 0.f);
  if (k < kL) {
    float2 s0, s1;
    if (k < kFB) {
      s0 = Hs[(size_t)(c * 2 + 0) * kFB + k];
      s1 = Hs[(size_t)(c * 2 + 1) * kFB + k];
    } else {
      int mm = kL - k;
      s0 = Hs[(size_t)(c * 2 + 0) * kFB + mm]; s0.y = -s0.y;
      s1 = Hs[(size_t)(c * 2 + 1) * kFB + mm]; s1.y = -s1.y;
    }
    float2 P = make_float2(s0.x - s1.y, s0.y + s1.x);  // S0 + i*S1
    double t = (double)k * (double)k / (double)(2 * kL);
    t -= floor(t);
    double sn, cs; sincos(TWO_PI_D * t, &sn, &cs);
    v = cmulf(P, make_float2((float)cs, (float)sn));
  }
  a[(size_t)c * kNB + k] = v;
}

__global__ void k_blu_mul(const float2* __restrict__ a, const float2* __restrict__ bb,
                          float2* __restrict__ q) {
  int k = blockIdx.x * blockDim.x + threadIdx.x;
  int c = blockIdx.y;
  if (k >= kNB) return;
  q[(size_t)c * kNB + k] = cmulf(a[(size_t)c * kNB + k], bb[k]);
}

// h_packed[c][n] = (1/(L*NB)) * e^{+i pi n^2/L} * q[c][n]  (re = h[c][0], im = h[c][1])
__global__ void k_makeh(const float2* __restrict__ q, float2* __restrict__ hC) {
  int n = blockIdx.x * blockDim.x + threadIdx.x;
  int c = blockIdx.y;
  if (n >= kNCV) return;
  float2 v = make_float2(0.f, 0.f);
  if (n < kL) {
    double t = (double)n * (double)n / (double)(2 * kL);
    t -= floor(t);
    double sn, cs; sincos(TWO_PI_D * t, &sn, &cs);
    float2 p = cmulf(q[(size_t)c * kNB + n], make_float2((float)cs, (float)sn));
    const float sc = 1.f / ((float)kL * (float)kNB);
    v = make_float2(p.x * sc, p.y * sc);
  }
  hC[(size_t)c * kNCV + n] = v;
}

// ================= convolution pipeline =================
__global__ void k_packx(const float* __restrict__ x, float2* __restrict__ A) {
  int n = blockIdx.x * blockDim.x + threadIdx.x;
  int b = blockIdx.y;
  if (n >= kNCV) return;
  float2 v = make_float2(0.f, 0.f);
  if (n < kT) {
    v.x = x[(size_t)(b * 2 + 0) * kT + n];
    v.y = x[(size_t)(b * 2 + 1) * kT + n];
  }
  A[(size_t)b * kNCV + n] = v;
}

// Unpack Hermitian halves of packed spectra, 2x2 MIMO combine, repack.
__global__ void k_combine(const float2* __restrict__ A, const float2* __restrict__ C,
                          float2* __restrict__ B) {
  int k = blockIdx.x * blockDim.x + threadIdx.x;
  int b = blockIdx.y;
  if (k >= kNCV) return;
  int kc = (kNCV - k) & (kNCV - 1);
  float2 Z  = A[(size_t)b * kNCV + k];
  float2 Zc = A[(size_t)b * kNCV + kc]; Zc.y = -Zc.y;
  float2 X0 = make_float2(0.5f * (Z.x + Zc.x), 0.5f * (Z.y + Zc.y));
  float2 X1 = make_float2(0.5f * (Z.y - Zc.y), -0.5f * (Z.x - Zc.x));
  float2 Y[2];
#pragma unroll
  for (int c = 0; c < 2; ++c) {
    float2 Gk  = C[(size_t)c * kNCV + k];
    float2 Gkc = C[(size_t)c * kNCV + kc]; Gkc.y = -Gkc.y;
    float2 H0 = make_float2(0.5f * (Gk.x + Gkc.x), 0.5f * (Gk.y + Gkc.y));
    float2 H1 = make_float2(0.5f * (Gk.y - Gkc.y), -0.5f * (Gk.x - Gkc.x));
    float2 y0 = cmulf(X0, H0);
    float2 y1 = cmulf(X1, H1);
    Y[c] = make_float2(y0.x + y1.x, y0.y + y1.y);
  }
  B[(size_t)b * kNCV + k] = make_float2(Y[0].x - Y[1].y, Y[0].y + Y[1].x);
}

__global__ void k_out(const float2* __restrict__ B, float* __restrict__ out) {
  int t = blockIdx.x * blockDim.x + threadIdx.x;
  int b = blockIdx.y;
  if (t >= kT) return;
  const float invN = 1.f / (float)kNCV;
  float2 v = B[(size_t)b * kNCV + t];
  out[(size_t)(b * 2 + 0) * kT + t] = v.x * invN;
  out[(size_t)(b * 2 + 1) * kT + t] = v.y * invN;
}

// ================= host driver =================
extern "C" void kernel_launch(void* const* d_in, const int* in_sizes, int n_in,
                              void* d_out, int out_size, void* d_ws, size_t ws_size,
                              hipStream_t stream) {
  (void)in_sizes; (void)n_in; (void)out_size; (void)ws_size;
  const float* x     = (const float*)d_in[0];
  const float* bmat  = (const float*)d_in[1];
  const float* cmat  = (const float*)d_in[2];
  const float* U_raw = (const float*)d_in[3];
  const float* g_raw = (const float*)d_in[4];
  float* out = (float*)d_out;

  char* w = (char*)d_ws;
  const size_t szBig = (size_t)kBAT * kNCV * sizeof(float2);   // 128 MB
  float2* bufA = (float2*)(w);
  float2* bufB = (float2*)(w + szBig);
  float2* bufC = (float2*)(w + 2 * szBig);
  float2* bufD = (float2*)(w + 2 * szBig + (size_t)2 * kNCV * sizeof(float2));
  // Bluestein scratch + params carved from bufB (dead before bufB is used as FFT pong)
  char* pb = (char*)bufB;
  float2* bluA  = (float2*)(pb);
  float2* bluA2 = (float2*)(pb + (size_t)2 * kNB * sizeof(float2));   // +4 MB
  float2* blub  = (float2*)(pb + ((size_t)8  << 20));
  float2* blub2 = (float2*)(pb + ((size_t)10 << 20));
  float2* Hs    = (float2*)(pb + ((size_t)12 << 20));
  float*  Umat  = (float*)(pb + ((size_t)14 << 20));
  float*  Gtab  = Umat + 64;

  // radix-4 Stockham; returns the buffer holding the result
  auto fft4 = [&](float2* buf, float2* tmp, int n, int stages, int batch,
                  float sign) -> float2* {
    float2* src = buf; float2* dst = tmp;
    int l = n >> 2, mshift = 0;
    for (int s = 0; s < stages; ++s) {
      dim3 grid((unsigned)((n / 4 + 255) / 256), (unsigned)batch);
      k_fft4_stage<<<grid, 256, 0, stream>>>(src, dst, l, mshift, sign, n);
      float2* t2 = src; src = dst; dst = t2;
      l >>= 2; mshift += 2;
    }
    return src;
  };

  // 1) parameters: expm (WMMA f32) + gamma table
  k_expm<<<1, 32, 0, stream>>>(U_raw, g_raw, Umat, Gtab);
  // 2) per-bin 6x6 complex solves -> H spectra
  k_solve<<<(kFB + 127) / 128, 128, 0, stream>>>(Umat, Gtab, bmat, cmat, Hs);
  // 3) Bluestein inverse DFT of length L (channels packed: 2 transforms)
  k_blu_b<<<kNB / 256, 256, 0, stream>>>(blub);
  float2* Fb = fft4(blub, blub2, kNB, kST_NB, 1, -1.f);         // 9 stages -> blub2
  k_blu_a<<<dim3(kNB / 256, 2), 256, 0, stream>>>(Hs, bluA);
  float2* Fa = fft4(bluA, bluA2, kNB, kST_NB, 2, -1.f);         // -> bluA2
  k_blu_mul<<<dim3(kNB / 256, 2), 256, 0, stream>>>(Fa, Fb, bluA);
  float2* Q = fft4(bluA, bluA2, kNB, kST_NB, 2, +1.f);          // -> bluA2
  // 4) chirp-demodulate -> packed h, zero-pad to conv FFT grid, forward FFT
  k_makeh<<<dim3(kNCV / 256, 2), 256, 0, stream>>>(Q, bufC);
  fft4(bufC, bufD, kNCV, kST_NCV, 2, -1.f);                     // 10 stages -> bufC
  // 5) pack x (2 real channels -> 1 complex), forward FFT in L2-resident groups
  k_packx<<<dim3(kNCV / 256, kBAT), 256, 0, stream>>>(x, bufA);
  for (int g2 = 0; g2 < kBAT / kGRP; ++g2)
    fft4(bufA + (size_t)g2 * kGRP * kNCV, bufB + (size_t)g2 * kGRP * kNCV,
         kNCV, kST_NCV, kGRP, -1.f);                            // -> bufA slices
  // 6) Hermitian unpack + 2x2 MIMO combine + repack
  k_combine<<<dim3(kNCV / 256, kBAT), 256, 0, stream>>>(bufA, bufC, bufB);
  // 7) inverse FFT (grouped), emit both output channels
  for (int g2 = 0; g2 < kBAT / kGRP; ++g2)
    fft4(bufB + (size_t)g2 * kGRP * kNCV, bufA + (size_t)g2 * kGRP * kNCV,
         kNCV, kST_NCV, kGRP, +1.f);                            // -> bufB slices
  k_out<<<dim3((kT + 255) / 256, kBAT), 256, 0, stream>>>(bufB, out);
}